// CRF_13280038880120
// MI455X (gfx1250) — compile-verified
//
#include <hip/hip_runtime.h>
#include <cstddef>

#define BTOT 512
#define TLEN 1024
#define LAB  48
#define BT   16
#define NTH  288          // 9 waves of 32: wave0 = WMMA/LSE, waves 1..8 = Viterbi
#define ALS  49           // padded stride for LSE alpha rows (bank-conflict free)
#define NEG_INF (-3.402823466e38f)

typedef __attribute__((ext_vector_type(2))) float v2f;
typedef __attribute__((ext_vector_type(8))) float v8f;

// ---------------------------------------------------------------------------
// Kernel 1: fused Viterbi forward + log-norm forward (WMMA) + sequence score.
// One block handles 16 batch rows for all 1024 time steps.
// ---------------------------------------------------------------------------
__global__ __launch_bounds__(NTH) void crf_fwd(
    const float* __restrict__ logits,    // [B,T,L]
    const int*   __restrict__ seq_lens,  // [B]
    const int*   __restrict__ labels,    // [B,T]
    const float* __restrict__ trans,     // [L,L]
    float*       __restrict__ out_ll,    // [B]
    unsigned int* __restrict__ bpw,      // [T,B,16] packed backpointer triples
    int*         __restrict__ lastt)     // [B]
{
  __shared__ float sTrans[LAB*LAB];
  __shared__ float sE[LAB*LAB];        // exp(trans)
  __shared__ float sAV[2][BT*LAB];     // viterbi alpha, double buffered
  __shared__ float sAL[BT*ALS];        // lse alpha (wave0 private across steps)
  __shared__ float sLg[BT*LAB];        // logits tile for step t
  __shared__ float sRM[BT];            // row maxes for lse
  __shared__ int   sLen[BT];

  const int tid = threadIdx.x;
  const int b0  = blockIdx.x * BT;

  // stage trans and exp(trans)
  for (int idx = tid; idx < LAB*LAB; idx += NTH) {
    float tv = trans[idx];
    sTrans[idx] = tv;
    sE[idx]     = __expf(tv);
  }
  if (tid < BT) sLen[tid] = seq_lens[b0 + tid];

  const bool isW0  = (tid < 32);
  const int  vl    = tid - 32;         // viterbi lane id (waves 1..8)
  const int  vb    = vl >> 4;          // batch row 0..15
  const int  j0    = 3 * (vl & 15);    // this lane owns labels j0..j0+2
  const bool isVit = (tid >= 32);
  const bool isSc  = isVit && ((vl & 15) == 0);  // 16 score lanes (one per b)

  const float* myLog = nullptr;
  float cur0=0.f, cur1=0.f, cur2=0.f;  // logits(t) for my 3 labels
  float sc_acc = 0.f;
  int lp = 0;

  if (isVit) {
    myLog = logits + ((size_t)(b0 + vb) * TLEN) * LAB + j0;
    // alpha(0) = logits[:,0,:]
    float a0 = myLog[0], a1 = myLog[1], a2 = myLog[2];
    int base  = vb*LAB + j0;
    sAV[0][base] = a0; sAV[0][base+1] = a1; sAV[0][base+2] = a2;
    int baseL = vb*ALS + j0;
    sAL[baseL] = a0; sAL[baseL+1] = a1; sAL[baseL+2] = a2;
    // prefetch logits(t=1)
    cur0 = myLog[LAB]; cur1 = myLog[LAB+1]; cur2 = myLog[LAB+2];
    if (isSc) {
      lp = labels[(size_t)(b0+vb)*TLEN];                       // label at t=0
      sc_acc = logits[((size_t)(b0+vb)*TLEN)*LAB + lp];        // unary t=0 (len>=1)
    }
  }
  __syncthreads();

  // wave 0: preload exp(trans) B-fragments into registers (constant all steps)
  v2f eb[36];
  int slenr[8];
  if (isW0) {
    const int n = tid & 15, khi = tid >> 4;
    #pragma unroll
    for (int nt = 0; nt < 3; ++nt)
      #pragma unroll
      for (int kc = 0; kc < 12; ++kc) {
        int k = 4*kc + 2*khi;
        v2f e;
        e.x = sE[(k  )*LAB + 16*nt + n];
        e.y = sE[(k+1)*LAB + 16*nt + n];
        eb[nt*12 + kc] = e;
      }
    #pragma unroll
    for (int r = 0; r < 8; ++r) slenr[r] = sLen[8*khi + r];
  }
  const int myLen = isVit ? sLen[vb] : 0;

  float nxt0=0.f, nxt1=0.f, nxt2=0.f;
  for (int t = 1; t < TLEN; ++t) {
    int lc = 0;
    if (isVit) {
      int base = vb*LAB + j0;
      sLg[base] = cur0; sLg[base+1] = cur1; sLg[base+2] = cur2;
      if (t + 1 < TLEN) {
        const float* p = myLog + (size_t)(t+1)*LAB;
        nxt0 = p[0]; nxt1 = p[1]; nxt2 = p[2];
        if (t + 2 < TLEN) __builtin_prefetch(myLog + (size_t)(t+2)*LAB, 0, 0);
      }
      if (isSc) lc = labels[(size_t)(b0+vb)*TLEN + t];
    }
    __syncthreads();   // sLg(t) ready; sAV[(t-1)&1] ready

    const int rd = (t-1) & 1, wr = t & 1;
    if (isVit) {
      if (t < myLen) {
        float m0=NEG_INF, m1=NEG_INF, m2=NEG_INF;
        int   p0=0, p1=0, p2=0;
        const float* av = &sAV[rd][vb*LAB];
        #pragma unroll 4
        for (int i = 0; i < LAB; ++i) {
          float a = av[i];
          const float* tr = &sTrans[i*LAB + j0];
          float s0 = a + tr[0], s1 = a + tr[1], s2 = a + tr[2];
          if (s0 > m0) { m0 = s0; p0 = i; }
          if (s1 > m1) { m1 = s1; p1 = i; }
          if (s2 > m2) { m2 = s2; p2 = i; }
        }
        int base = vb*LAB + j0;
        sAV[wr][base]   = m0 + cur0;
        sAV[wr][base+1] = m1 + cur1;
        sAV[wr][base+2] = m2 + cur2;
        bpw[((size_t)t*BTOT + (b0+vb))*16 + (vl & 15)] =
            (unsigned)p0 | ((unsigned)p1 << 8) | ((unsigned)p2 << 16);
      } else {
        int base = vb*LAB + j0;     // frozen: carry alpha forward
        sAV[wr][base]   = sAV[rd][base];
        sAV[wr][base+1] = sAV[rd][base+1];
        sAV[wr][base+2] = sAV[rd][base+2];
      }
      if (isSc) {
        if (t < myLen) sc_acc += sLg[vb*LAB + lc] + sTrans[lp*LAB + lc];
        lp = lc;
      }
    } else {
      // ---- wave 0: log-sum-exp step via f32 WMMA: S = exp(alpha-m) @ exp(trans)
      const int n = tid & 15, khi = tid >> 4;
      float mm = NEG_INF;
      #pragma unroll 8
      for (int i = 0; i < LAB; ++i) mm = fmaxf(mm, sAL[n*ALS + i]);
      sRM[n] = mm;                       // (lanes 16..31 duplicate, benign)
      v8f c0 = {}, c1 = {}, c2 = {};
      #pragma unroll
      for (int kc = 0; kc < 12; ++kc) {
        int k = 4*kc + 2*khi;
        v2f a;
        a.x = __expf(sAL[n*ALS + k    ] - mm);
        a.y = __expf(sAL[n*ALS + k + 1] - mm);
        c0 = __builtin_amdgcn_wmma_f32_16x16x4_f32(false, a, false, eb[kc],      (short)0, c0, false, false);
        c1 = __builtin_amdgcn_wmma_f32_16x16x4_f32(false, a, false, eb[12 + kc], (short)0, c1, false, false);
        c2 = __builtin_amdgcn_wmma_f32_16x16x4_f32(false, a, false, eb[24 + kc], (short)0, c2, false, false);
      }
      asm volatile("s_wait_dscnt 0" ::: "memory");  // sRM stores visible wave-wide
      #pragma unroll
      for (int nt = 0; nt < 3; ++nt) {
        v8f c = (nt == 0) ? c0 : (nt == 1) ? c1 : c2;
        #pragma unroll
        for (int r = 0; r < 8; ++r) {
          int m = 8*khi + r;
          int j = 16*nt + n;
          float na = sRM[m] + __logf(c[r]) + sLg[m*LAB + j];
          float oa = sAL[m*ALS + j];
          sAL[m*ALS + j] = (t < slenr[r]) ? na : oa;
        }
      }
    }
    __syncthreads();   // all reads of sAV[rd]/sLg done before next overwrite
    cur0 = nxt0; cur1 = nxt1; cur2 = nxt2;
  }

  // epilogue: log-norm, log-likelihood, last Viterbi tag (16 score lanes)
  if (isSc) {
    const int b = vb;
    float m = NEG_INF;
    #pragma unroll 8
    for (int j = 0; j < LAB; ++j) m = fmaxf(m, sAL[b*ALS + j]);
    float s = 0.f;
    #pragma unroll 8
    for (int j = 0; j < LAB; ++j) s += __expf(sAL[b*ALS + j] - m);
    float lognorm = m + __logf(s);
    out_ll[b0 + b] = sc_acc - lognorm;

    float mv = NEG_INF; int bt = 0;
    const float* av = &sAV[(TLEN-1) & 1][b*LAB];
    #pragma unroll 8
    for (int j = 0; j < LAB; ++j) {
      float v = av[j];
      if (v > mv) { mv = v; bt = j; }
    }
    lastt[b0 + b] = bt;
  }
}

// ---------------------------------------------------------------------------
// Kernel 2: Viterbi backtrace. One block per batch row; backpointer rows are
// staged through LDS in 256-step chunks (double buffered) so the serial
// pointer chase runs at LDS latency.
// ---------------------------------------------------------------------------
__global__ __launch_bounds__(256) void crf_back(
    const unsigned int* __restrict__ bpw,
    const int* __restrict__ seq_lens,
    const int* __restrict__ lastt,
    float* __restrict__ out_tags)     // [B,T]
{
  __shared__ unsigned int  sBp[2][256*16];
  __shared__ unsigned char sTag[256];
  const int b   = blockIdx.x;
  const int tid = threadIdx.x;
  const int len = seq_lens[b];

  uint4 rg[4];
  // load chunk 3 into sBp[0]
  #pragma unroll
  for (int q = 0; q < 4; ++q) {
    int idx = tid + 256*q, tl = idx >> 2, part = idx & 3;
    rg[q] = *(const uint4*)(bpw + ((size_t)(3*256 + tl)*BTOT + b)*16 + part*4);
  }
  #pragma unroll
  for (int q = 0; q < 4; ++q) {
    int idx = tid + 256*q, tl = idx >> 2, part = idx & 3;
    *(uint4*)(&sBp[0][tl*16 + part*4]) = rg[q];
  }
  __syncthreads();

  int cur = lastt[b];
  int buf = 0;
  for (int c = 3; c >= 0; --c) {
    if (c > 0) {           // prefetch next-lower chunk while walking this one
      #pragma unroll
      for (int q = 0; q < 4; ++q) {
        int idx = tid + 256*q, tl = idx >> 2, part = idx & 3;
        rg[q] = *(const uint4*)(bpw + ((size_t)((c-1)*256 + tl)*BTOT + b)*16 + part*4);
      }
    }
    if (tid == 0) {
      const int lo = c*256;
      int cc = cur;
      for (int tt = lo + 255; tt >= (lo > 0 ? lo : 1); --tt) {
        sTag[tt - lo] = (unsigned char)cc;
        if (tt < len) {
          unsigned w = sBp[buf][(tt - lo)*16 + cc/3];
          cc = (int)((w >> (8*(cc % 3))) & 63u);
        }
      }
      if (c == 0) sTag[0] = (unsigned char)cc;
      cur = cc;
    }
    __syncthreads();
    int pos = c*256 + tid;
    out_tags[(size_t)b*TLEN + pos] = (pos < len) ? (float)sTag[tid] : 0.0f;
    if (c > 0) {
      #pragma unroll
      for (int q = 0; q < 4; ++q) {
        int idx = tid + 256*q, tl = idx >> 2, part = idx & 3;
        *(uint4*)(&sBp[buf ^ 1][tl*16 + part*4]) = rg[q];
      }
      __syncthreads();
      buf ^= 1;
    }
  }
}

// ---------------------------------------------------------------------------
extern "C" void kernel_launch(void* const* d_in, const int* in_sizes, int n_in,
                              void* d_out, int out_size, void* d_ws, size_t ws_size,
                              hipStream_t stream) {
  const float* logits = (const float*)d_in[0];
  const int*   slen   = (const int*)d_in[1];
  const int*   labels = (const int*)d_in[2];
  const float* trans  = (const float*)d_in[3];
  float* out = (float*)d_out;

  unsigned int* bpw = (unsigned int*)d_ws;                        // 32 MB
  int* lastt = (int*)((char*)d_ws + (size_t)TLEN * BTOT * 16 * 4); // +2 KB

  crf_fwd<<<BTOT / BT, NTH, 0, stream>>>(logits, slen, labels, trans,
                                         out, bpw, lastt);
  crf_back<<<BTOT, 256, 0, stream>>>(bpw, slen, lastt, out + BTOT);

  (void)in_sizes; (void)n_in; (void)out_size; (void)ws_size;
}